// rModule_73151882985840
// MI455X (gfx1250) — compile-verified
//
#include <hip/hip_runtime.h>
#include <hip/hip_bf16.h>
#include <stdint.h>

// Problem constants (fixed by the reference).
#define Bn 8
#define Cn 3
#define Hn 1024
#define Wn 1024
#define PLANES (Bn * Cn)                 // 24
#define PLANE_ELEMS ((size_t)Hn * Wn)    // 1M

#define TILE_H 16
#define TILE_W 64
// Staged window spans [w0-4, w0+67]: 18 aligned groups of 4 floats.
#define GROUPS 18
#define LDS_STRIDE 72                    // floats; 288 B, 16B-aligned rows
#define MK_ROWS (TILE_H + 2)             // 18: rows h0-1 .. h0+16
#define LK_ROWS (TILE_H + 1)             // 17: rows h0   .. h0+16
#define MK_TASKS (MK_ROWS * GROUPS)      // 324
#define LK_TASKS (LK_ROWS * GROUPS)      // 306

typedef __attribute__((ext_vector_type(4))) float v4f;

// Async global -> LDS copy, 16 bytes per active lane (tracked by ASYNCcnt).
// VDST = LDS byte offset (relative to wave LDS base), VADDR = 64-bit global
// address, saddr = off mode. Both addresses are 16B-aligned by construction.
__device__ __forceinline__ void async_copy_b128(unsigned lds_byte_off,
                                                const float* gptr) {
    asm volatile("global_load_async_to_lds_b128 %0, %1, off"
                 :
                 : "v"(lds_byte_off),
                   "v"((unsigned long long)(uintptr_t)gptr)
                 : "memory");
}

__device__ __forceinline__ void wait_async_all() {
    asm volatile("s_wait_asynccnt 0" ::: "memory");
}

__global__ __launch_bounds__(256)
void mr_stencil_kernel(const float* __restrict__ dA,
                       const float* __restrict__ aty,
                       const float* __restrict__ mk,
                       const float* __restrict__ lk,
                       const float* __restrict__ gnp,
                       const float* __restrict__ gpp,
                       const int*   __restrict__ scp,
                       float* __restrict__ out_m,
                       float* __restrict__ out_s) {
    __shared__ float s_mk[MK_ROWS * LDS_STRIDE];
    __shared__ float s_lk[LK_ROWS * LDS_STRIDE];

    const int tid = threadIdx.x;                 // block = 256 flat
    const int w0 = blockIdx.x * TILE_W;
    const int h0 = blockIdx.y * TILE_H;
    const int plane = blockIdx.z;

    const size_t pbase = (size_t)plane * PLANE_ELEMS;
    const float* mkp = mk + pbase;
    const float* lkp = lk + pbase;

    // LDS byte offsets of the staging buffers (flat shared ptr low 32 bits
    // == LDS offset on gfx1250).
    const unsigned mk_base = (unsigned)(uintptr_t)(&s_mk[0]);
    const unsigned lk_base = (unsigned)(uintptr_t)(&s_lk[0]);

    // ---- Stage mk rows h0-1..h0+16 over cols [w0-4, w0+67] (wrapped). ----
    // Each 4-float group stays contiguous under wrap: (w0-4+4g)&1023 is
    // 4-aligned and never crosses a row end.
    for (int t = tid; t < MK_TASKS; t += 256) {
        const int r = t / GROUPS;
        const int g = t - r * GROUPS;
        const int gh = (h0 + r - 1) & (Hn - 1);
        const int gw = (w0 - 4 + 4 * g) & (Wn - 1);
        async_copy_b128(mk_base + (unsigned)(r * LDS_STRIDE + 4 * g) * 4u,
                        mkp + (size_t)gh * Wn + gw);
    }
    // ---- Stage lambdak rows h0..h0+16 over the same window. ----
    for (int t = tid; t < LK_TASKS; t += 256) {
        const int r = t / GROUPS;
        const int g = t - r * GROUPS;
        const int gh = (h0 + r) & (Hn - 1);
        const int gw = (w0 - 4 + 4 * g) & (Wn - 1);
        async_copy_b128(lk_base + (unsigned)(r * LDS_STRIDE + 4 * g) * 4u,
                        lkp + (size_t)gh * Wn + gw);
    }

    wait_async_all();      // this wave's async copies have landed in LDS
    __syncthreads();       // all waves' copies visible

    const float gn = gnp[0];
    const float gp = gpp[0];
    const int   sc = scp[0];

    // One row x 4 consecutive columns per thread.
    const int y  = tid >> 4;               // 0..15
    const int x0 = (tid & 15) * 4;         // 0,4,...,60
    const int cl = x0 + 4;                 // LDS col of global (w0 + x0)

    const int gh = h0 + y;
    const size_t gidx = pbase + (size_t)gh * Wn + (w0 + x0);

    // Streaming inputs: single use -> nontemporal, keep L2 for mk/lk halos.
    const v4f dAv = __builtin_nontemporal_load((const v4f*)(dA + gidx));
    const v4f av  = __builtin_nontemporal_load((const v4f*)(aty + gidx));

    // Vector LDS reads (all 16B aligned with stride 72).
    const v4f mCv = *(const v4f*)&s_mk[(y + 1) * LDS_STRIDE + cl];
    const v4f mUv = *(const v4f*)&s_mk[(y + 0) * LDS_STRIDE + cl];
    const v4f mDv = *(const v4f*)&s_mk[(y + 2) * LDS_STRIDE + cl];
    const v4f lCv = *(const v4f*)&s_lk[(y + 0) * LDS_STRIDE + cl];
    const v4f lDv = *(const v4f*)&s_lk[(y + 1) * LDS_STRIDE + cl];
    // Scalar edge reads for the shifted-by-one neighbors.
    const float mLs = s_mk[(y + 1) * LDS_STRIDE + (cl - 1)];
    const float mRs = s_mk[(y + 1) * LDS_STRIDE + (cl + 4)];
    const float lRs = s_lk[(y + 0) * LDS_STRIDE + (cl + 4)];

    v4f mrv, sgv;
#pragma unroll
    for (int i = 0; i < 4; ++i) {
        const float mC = mCv[i];
        const float mL = (i == 0) ? mLs : mCv[i - 1];
        const float mR = (i == 3) ? mRs : mCv[i + 1];
        const float mU = mUv[i];
        const float mD = mDv[i];
        const float lC = lCv[i];
        const float lR = (i == 3) ? lRs : lCv[i + 1];
        const float lD = lDv[i];

        // diagDtlambdakD = 2*lk + lk(w+1) + lk(h+1)
        const float diagD = 2.0f * lC + lR + lD;

        // invSigma = relu(gn*dA + gp*diagD) + 1e-4
        float sig = gn * dAv[i] + gp * diagD;
        sig = fmaxf(sig, 0.0f) + 1.0e-4f;

        // AtAmk: mk at (h%scale==0 && w%scale==0); fast path for scale==2.
        const int gw = w0 + x0 + i;
        const bool keep = (sc == 2) ? (((gh | gw) & 1) == 0)
                                    : ((gh % sc) == 0 && (gw % sc) == 0);
        const float atamk = keep ? mC : 0.0f;

        // Dht(lk*Dh(mk)) and Dvt(lk*Dv(mk)) with circular wrap (LDS halo).
        const float dht = lC * (mC - mL) - lR * (mR - mC);
        const float dvt = lC * (mC - mU) - lD * (mD - mC);

        const float num = gn * (av[i] - atamk + dAv[i] * mC)
                        - gp * (dht + dvt)
                        + gp * diagD * mC;
        mrv[i] = num / sig;
        sgv[i] = sig;
    }

    // Outputs are never re-read: nontemporal stores.
    __builtin_nontemporal_store(mrv, (v4f*)(out_m + gidx));
    __builtin_nontemporal_store(sgv, (v4f*)(out_s + gidx));
}

extern "C" void kernel_launch(void* const* d_in, const int* in_sizes, int n_in,
                              void* d_out, int out_size, void* d_ws, size_t ws_size,
                              hipStream_t stream) {
    (void)in_sizes; (void)n_in; (void)out_size; (void)d_ws; (void)ws_size;

    const float* dA  = (const float*)d_in[0];   // diagAtA
    const float* aty = (const float*)d_in[1];   // Aty
    const float* mk  = (const float*)d_in[2];   // mk
    const float* lk  = (const float*)d_in[3];   // lambdak
    const float* gn  = (const float*)d_in[4];   // gamma_n (1 elem)
    const float* gp  = (const float*)d_in[5];   // gamma_p (1 elem)
    const int*   sc  = (const int*)d_in[6];     // scale   (1 elem)

    const size_t N = (size_t)PLANES * PLANE_ELEMS;   // 25,165,824
    float* out_m = (float*)d_out;       // m_r        (first output)
    float* out_s = out_m + N;           // invSigma_r (second output)

    dim3 block(256, 1, 1);                          // 8 wave32s
    dim3 grid(Wn / TILE_W, Hn / TILE_H, PLANES);    // 16 x 64 x 24

    mr_stencil_kernel<<<grid, block, 0, stream>>>(dA, aty, mk, lk, gn, gp, sc,
                                                  out_m, out_s);
}